// Mol_substruct_Experts_85177791414565
// MI455X (gfx1250) — compile-verified
//
#include <hip/hip_runtime.h>
#include <hip/hip_bf16.h>

// Problem constants (match reference): S=4096, B=2048, L=256, D=128
#define NB   2048
#define NL   256
#define ND   128

typedef __attribute__((ext_vector_type(2))) float v2f;
typedef __attribute__((ext_vector_type(8))) float v8f;

// ---------------------------------------------------------------------------
// Kernel A: MT[k][n] = dot(Wq[k,:], Wk[n,:])   (= (Wq @ Wk^T), 128x128)
//           vvec[n]  = dot(Wk[n,:], bq)        (bias fold-in for u)
// Tiny (4 MFLOP); single workgroup.
// ---------------------------------------------------------------------------
__global__ void precompute_kernel(const float* __restrict__ Wq,
                                  const float* __restrict__ Wk,
                                  const float* __restrict__ bq,
                                  float* __restrict__ MT,
                                  float* __restrict__ vvec) {
    int t = threadIdx.x;
    for (int e = t; e < ND * ND; e += 256) {
        int k = e >> 7, n = e & (ND - 1);
        const float* wq = Wq + k * ND;
        const float* wk = Wk + n * ND;
        float acc = 0.f;
#pragma unroll 8
        for (int d = 0; d < ND; ++d) acc += wq[d] * wk[d];
        MT[e] = acc;
    }
    for (int n = t; n < ND; n += 256) {
        const float* wk = Wk + n * ND;
        float acc = 0.f;
#pragma unroll 8
        for (int d = 0; d < ND; ++d) acc += wk[d] * bq[d];
        vvec[n] = acc;
    }
}

// ---------------------------------------------------------------------------
// Kernel B: U = G0 @ MT via V_WMMA_F32_16X16X4_F32 (f32, matches reference
// precision). G0[b] = sub_feat[sub_index[b*L + 0]].
// Grid: NB/16 blocks of 256 threads (8 waves). Each wave owns one 16x16
// N-tile (8 waves * 16 = 128 = D); K loop = 128 in steps of 4 -> 32 WMMA/wave.
//
// A (16x4 f32) lane layout: lanes 0-15 row M=lane, VGPR0/1 = K{0,1};
//                           lanes 16-31 row M=lane-16, VGPR0/1 = K{2,3}.
// B (4x16 f32): VGPR0 = row K0 (lanes 0-15) / K2 (lanes 16-31), VGPR1 = K1/K3.
// D (16x16 f32, 8 VGPRs): VGPR r -> (M=r, N=lane) / (M=r+8, N=lane-16).
// ---------------------------------------------------------------------------
__global__ void u_gemm_kernel(const float* __restrict__ sub_feat,
                              const int* __restrict__ sub_index,
                              const float* __restrict__ MT,
                              float* __restrict__ U) {
    // Stride 132 -> conflict-free ds_load_b64 column reads, 8B-aligned rows.
    __shared__ float As[16][132];

    const int b0   = blockIdx.x * 16;
    const int t    = threadIdx.x;
    const int lane = t & 31;
    const int wave = t >> 5;      // 0..7
    const int n0   = wave * 16;   // N-tile base for this wave

    // Stage A-tile: 16 gathered feature rows (position 0 of each batch).
    {
        int r = t >> 4;            // 0..15
        int c = (t & 15) * 8;      // 0..120
        int idx = sub_index[(b0 + r) * NL];     // sub_index[b][0]
        const float4* src = (const float4*)(sub_feat + (long)idx * ND + c);
        float4 x0 = src[0];
        float4 x1 = src[1];
        float* dst = &As[r][c];
        *(v2f*)(dst + 0) = (v2f){x0.x, x0.y};
        *(v2f*)(dst + 2) = (v2f){x0.z, x0.w};
        *(v2f*)(dst + 4) = (v2f){x1.x, x1.y};
        *(v2f*)(dst + 6) = (v2f){x1.z, x1.w};
    }
    __syncthreads();

    const int row = lane & 15;            // A row for this lane
    const int col = lane & 15;            // B/D column
    const int kb  = (lane >> 4) * 2;      // K sub-pair select

    v8f acc = {};
#pragma unroll 8
    for (int k0 = 0; k0 < ND; k0 += 4) {
        int kk = k0 + kb;
        v2f a = *(const v2f*)&As[row][kk];           // A[row][kk], A[row][kk+1]
        v2f b;
        b.x = MT[kk * ND + n0 + col];                // B[kk][col]   = MT[kk][n0+col]
        b.y = MT[(kk + 1) * ND + n0 + col];          // B[kk+1][col]
        acc = __builtin_amdgcn_wmma_f32_16x16x4_f32(
            /*neg_a=*/false, a, /*neg_b=*/false, b,
            /*c_mod=*/(short)0, acc, /*reuse_a=*/false, /*reuse_b=*/false);
    }

    // Scatter D tile: VGPR r -> row (r + 8*(lane>>4)), col = lane&15.
    const int rbase = (lane >> 4) * 8;
#pragma unroll
    for (int r = 0; r < 8; ++r) {
        U[(b0 + rbase + r) * ND + n0 + col] = acc[r];
    }
}

// ---------------------------------------------------------------------------
// Kernel C: per-batch scores + softmax + weighted gather-sum.
//   u_b   = U[b] + vvec
//   s_j   = dot(g_j, u_b)/sqrt(D) + (1-mask)*-1e4
//   p     = softmax(s); out[b] = sum_j p_j * g_j
// One workgroup (256 threads) per batch; all gathers hit L2 (sub_feat = 2MB).
// ---------------------------------------------------------------------------
__global__ void attn_kernel(const float* __restrict__ sub_feat,
                            const int* __restrict__ sub_index,
                            const float* __restrict__ mask,
                            const float* __restrict__ U,
                            const float* __restrict__ vvec,
                            float* __restrict__ out) {
    __shared__ float u[ND];
    __shared__ float sc[NL];
    __shared__ float p[NL];
    __shared__ int   ids[NL];
    __shared__ float part[ND];

    const int b = blockIdx.x;
    const int t = threadIdx.x;

    if (t < ND) u[t] = U[b * ND + t] + vvec[t];
    const int idx = sub_index[b * NL + t];
    ids[t] = idx;
    __syncthreads();

    // Pass 1: score for j = t
    const float* g = sub_feat + (long)idx * ND;
    float acc = 0.f;
#pragma unroll 8
    for (int d = 0; d < ND; d += 4) {
        float4 gv = *(const float4*)(g + d);
        acc += gv.x * u[d] + gv.y * u[d + 1] + gv.z * u[d + 2] + gv.w * u[d + 3];
    }
    const float inv_sqrt_d = 0.08838834764831845f;  // 1/sqrt(128)
    float s = acc * inv_sqrt_d + (1.0f - mask[b * NL + t]) * -10000.0f;

    // Softmax: tree max
    sc[t] = s; __syncthreads();
    for (int off = 128; off > 0; off >>= 1) {
        if (t < off) sc[t] = fmaxf(sc[t], sc[t + off]);
        __syncthreads();
    }
    const float mx = sc[0];
    __syncthreads();

    float e = __expf(s - mx);
    sc[t] = e; __syncthreads();
    for (int off = 128; off > 0; off >>= 1) {
        if (t < off) sc[t] += sc[t + off];
        __syncthreads();
    }
    p[t] = e * (1.0f / sc[0]);
    __syncthreads();

    // Pass 2: out[b][d] = sum_j p[j] * g_j[d]; split j-range over two halves.
    const int d    = t & (ND - 1);
    const int half = t >> 7;
    float o = 0.f;
    const int j0 = half * 128;
#pragma unroll 4
    for (int j = j0; j < j0 + 128; ++j) {
        o += p[j] * sub_feat[(long)ids[j] * ND + d];
    }
    if (half == 1) part[d] = o;
    __syncthreads();
    if (half == 0) out[b * ND + d] = o + part[d];
}

// ---------------------------------------------------------------------------
// Inputs (setup_inputs order):
//   0: sub_feat  (S*D f32)   1: mask (B*L f32)   2: Wq (D*D f32)
//   3: bq (D f32)            4: Wk (D*D f32)     5: bk (D f32)
//   6: sub_index (B*L int32)
// Output: drug_embeddings (B*D f32)
// Workspace: MT (16384 f32) | vvec (128 f32) | U (B*D f32)
// ---------------------------------------------------------------------------
extern "C" void kernel_launch(void* const* d_in, const int* in_sizes, int n_in,
                              void* d_out, int out_size, void* d_ws, size_t ws_size,
                              hipStream_t stream) {
    const float* sub_feat  = (const float*)d_in[0];
    const float* mask      = (const float*)d_in[1];
    const float* Wq        = (const float*)d_in[2];
    const float* bq        = (const float*)d_in[3];
    const float* Wk        = (const float*)d_in[4];
    // bk only affects a per-batch constant added to every score -> cancels in
    // softmax; intentionally unused.
    const int*   sub_index = (const int*)d_in[6];
    float*       out       = (float*)d_out;

    float* MT   = (float*)d_ws;                   // 128*128
    float* vvec = MT + ND * ND;                   // 128
    float* U    = vvec + ND;                      // NB*ND

    precompute_kernel<<<1, 256, 0, stream>>>(Wq, Wk, bq, MT, vvec);
    u_gemm_kernel<<<NB / 16, 256, 0, stream>>>(sub_feat, sub_index, MT, U);
    attn_kernel<<<NB, 256, 0, stream>>>(sub_feat, sub_index, mask, U, vvec, out);
}